// APPLY_PENALTY_5909874999394
// MI455X (gfx1250) — compile-verified
//
#include <hip/hip_runtime.h>

// Problem constants from the reference module.
#define B_ROWS 256
#define VOCAB 128000
#define HIST 512

typedef float v4f __attribute__((ext_vector_type(4)));

// ---------------------------------------------------------------------------
// Kernel 1: streaming copy logits -> out.
// 262 MB of HBM traffic => the whole op's cost. Use 128-bit accesses with
// non-temporal hints (gfx1250 TH=NT cache control) so the one-shot stream
// doesn't pollute WGP$/L2. Each thread moves 4 float4s (64 B), strided by
// blockDim within the block so every global_load_b128 is fully coalesced
// (32 lanes x 16 B = 512 B per instruction).
// Total float4s: 256*128000/4 = 8,192,000 = 8000 blocks * 1024 float4s.
// ---------------------------------------------------------------------------
__global__ __launch_bounds__(256) void APPLY_PENALTY_copy_kernel(
    const v4f* __restrict__ src, v4f* __restrict__ dst) {
  const size_t base = (size_t)blockIdx.x * 1024u + threadIdx.x;

  v4f a = __builtin_nontemporal_load(&src[base]);
  v4f b = __builtin_nontemporal_load(&src[base + 256]);
  v4f c = __builtin_nontemporal_load(&src[base + 512]);
  v4f d = __builtin_nontemporal_load(&src[base + 768]);

  __builtin_nontemporal_store(a, &dst[base]);
  __builtin_nontemporal_store(b, &dst[base + 256]);
  __builtin_nontemporal_store(c, &dst[base + 512]);
  __builtin_nontemporal_store(d, &dst[base + 768]);
}

// ---------------------------------------------------------------------------
// Kernel 2: apply penalty to the last `penalty_range` history slots per row.
// One thread per (row, history slot) over the full HIST width; threads whose
// slot is outside the active range exit. penalty_range is read from its
// device-side scalar (no host readback -> graph-capture safe).
// Duplicate indices are idempotent: every write is logits[row,idx] * p,
// matching jnp .set semantics. Reads come from the ORIGINAL logits input.
// 256*512 threads = 512 blocks; negligible vs. the copy.
// ---------------------------------------------------------------------------
__global__ __launch_bounds__(256) void APPLY_PENALTY_scatter_kernel(
    const float* __restrict__ logits,
    const int* __restrict__ save_id,
    const float* __restrict__ penalty_value,
    const int* __restrict__ penalty_range,
    float* __restrict__ out) {
  const int gid = blockIdx.x * blockDim.x + threadIdx.x;  // 0 .. B*HIST-1
  const int row = gid / HIST;
  const int j   = gid - row * HIST;                       // history slot

  const int range = penalty_range[0];                     // 64 in reference
  const int start = HIST - range;
  if (j < start) return;

  const float p = penalty_value[0];
  const int idx = save_id[(size_t)row * HIST + j];        // 0 .. VOCAB-1
  const size_t off = (size_t)row * VOCAB + (size_t)idx;
  out[off] = logits[off] * p;
}

extern "C" void kernel_launch(void* const* d_in, const int* in_sizes, int n_in,
                              void* d_out, int out_size, void* d_ws, size_t ws_size,
                              hipStream_t stream) {
  (void)in_sizes; (void)n_in; (void)out_size; (void)d_ws; (void)ws_size;

  const float* logits        = (const float*)d_in[0];  // (256, 128000) fp32
  const int*   save_id       = (const int*)d_in[1];    // (256, 512) int
  const float* penalty_value = (const float*)d_in[2];  // scalar fp32
  const int*   penalty_range = (const int*)d_in[3];    // scalar int

  float* out = (float*)d_out;

  // Copy: 8,192,000 float4s / (256 threads * 4 float4s) = 8000 blocks.
  constexpr int kCopyBlocks = (B_ROWS * VOCAB / 4) / 1024;  // == 8000 exactly
  APPLY_PENALTY_copy_kernel<<<kCopyBlocks, 256, 0, stream>>>(
      (const v4f*)logits, (v4f*)out);

  // Scatter: one thread per (row, history slot); stream order gives the
  // copy -> scatter dependency.
  constexpr int kScatterBlocks = (B_ROWS * HIST) / 256;     // == 512
  APPLY_PENALTY_scatter_kernel<<<kScatterBlocks, 256, 0, stream>>>(
      logits, save_id, penalty_value, penalty_range, out);
}